// GraphEncoder_51771535786305
// MI455X (gfx1250) — compile-verified
//
#include <hip/hip_runtime.h>

#define N_NODES 100000
#define N_EDGES 1200000
#define IN_FEATS 30
#define HIDDEN 64

typedef __attribute__((ext_vector_type(2))) float v2f;
typedef __attribute__((ext_vector_type(8))) float v8f;

// ---------------------------------------------------------------- utilities
__global__ void zero_f32_kernel(float* __restrict__ p, int n) {
    int i = blockIdx.x * blockDim.x + threadIdx.x;
    if (i < n) p[i] = 0.0f;
}

__global__ void degree_kernel(const int* __restrict__ src, const int* __restrict__ dst,
                              float* __restrict__ out_deg, float* __restrict__ in_deg,
                              int n_edges) {
    int e = blockIdx.x * blockDim.x + threadIdx.x;
    if (e >= n_edges) return;
    atomicAdd(&out_deg[src[e]], 1.0f);
    atomicAdd(&in_deg[dst[e]], 1.0f);
}

__global__ void norm_finalize_kernel(float* __restrict__ deg, int n) {
    int i = blockIdx.x * blockDim.x + threadIdx.x;
    if (i < n) deg[i] = rsqrtf(fmaxf(deg[i], 1.0f));
}

// ------------------------------------------------- dense transform via WMMA
// H[N x 64] = (X * normA[:,None]) @ W,  W is [K x 64].
// One wave32 computes one 16x16 tile of H using V_WMMA_F32_16X16X4_F32.
//
// f32 16x16x4 VGPR layouts (ISA 7.12.2):
//   A (16x4): lane l -> M = l%16; a[0]=A[M][k0+kh], a[1]=A[M][k0+kh+1], kh = (l>=16)?2:0
//   B (4x16): lane l -> N = l%16; b[0]=B[k0+kh][N], b[1]=B[k0+kh+1][N]
//   C/D (16x16): VGPR r: lanes 0-15 -> (M=r, N=lane); lanes 16-31 -> (M=r+8, N=lane-16)
//
// K-tail handling is branch-free: out-of-range lanes load a clamped (valid)
// address and select 0.0f via v_cndmask, so EXEC never changes around WMMA.
template <int K>
__global__ void gemm_norm_wmma_kernel(const float* __restrict__ X,
                                      const float* __restrict__ W,
                                      const float* __restrict__ normA,
                                      float* __restrict__ H,
                                      int n_rows) {
    const int lane = threadIdx.x & 31;
    const int wave = threadIdx.x >> 5;
    const int tiles_n = HIDDEN / 16;                       // 4
    const int tile = blockIdx.x * (blockDim.x >> 5) + wave;
    const int tile_m = tile / tiles_n;
    const int tile_n = tile % tiles_n;
    if (tile_m * 16 >= n_rows) return;                     // uniform per wave

    const int half = lane >> 4;                            // 0 or 1
    const int m = tile_m * 16 + (lane & 15);
    const int n = tile_n * 16 + (lane & 15);
    const int kh = half << 1;                              // 0 or 2

    const float scale = normA[m];
    const float* __restrict__ xrow = X + (size_t)m * K;

    v8f acc = {};
    #pragma unroll
    for (int k0 = 0; k0 < K; k0 += 4) {
        const int ka = k0 + kh;
        const bool in0 = (ka < K);
        const bool in1 = (ka + 1 < K);
        const int ia0 = in0 ? ka : 0;                      // clamped, always valid
        const int ia1 = in1 ? ka + 1 : 0;
        const float xa0 = xrow[ia0];
        const float xa1 = xrow[ia1];
        const float wb0 = W[(size_t)ia0 * HIDDEN + n];
        const float wb1 = W[(size_t)ia1 * HIDDEN + n];
        v2f a, b;
        a[0] = in0 ? xa0 * scale : 0.0f;
        a[1] = in1 ? xa1 * scale : 0.0f;
        b[0] = in0 ? wb0 : 0.0f;
        b[1] = in1 ? wb1 : 0.0f;
        acc = __builtin_amdgcn_wmma_f32_16x16x4_f32(
            /*neg_a=*/false, a, /*neg_b=*/false, b,
            /*c_mod=*/(short)0, acc, /*reuse_a=*/false, /*reuse_b=*/false);
    }

    const int m_out = tile_m * 16 + (half << 3);           // +8 for upper half lanes
    float* __restrict__ hout = H + (size_t)m_out * HIDDEN + n;
    #pragma unroll
    for (int r = 0; r < 8; ++r)
        hout[(size_t)r * HIDDEN] = acc[r];
}

// ------------------------------------------------------- sparse aggregation
// agg[dst[e]][:] += h[src[e]][:].  One wave per edge (edge id uniform per
// wave -> single coalesced index fetch); each lane handles 2 features:
// coalesced 256B row read + 64 global_atomic_add_f32 into one dst row.
__global__ void edge_scatter_kernel(const float* __restrict__ h,
                                    const int* __restrict__ src,
                                    const int* __restrict__ dst,
                                    float* __restrict__ agg,
                                    int n_edges) {
    int t = blockIdx.x * blockDim.x + threadIdx.x;
    int e = t >> 5;                      // wave-uniform
    int f = (t & 31) << 1;               // 0,2,...,62
    if (e >= n_edges) return;
    int s = src[e];
    int d = dst[e];
    const float2 v = *reinterpret_cast<const float2*>(h + (size_t)s * HIDDEN + f);
    atomicAdd(&agg[(size_t)d * HIDDEN + f], v.x);
    atomicAdd(&agg[(size_t)d * HIDDEN + f + 1], v.y);
}

// out = relu(agg * norm_dst[:,None] + b)  (in-place safe: agg may alias out)
__global__ void bias_relu_kernel(const float* __restrict__ agg,
                                 const float* __restrict__ norm_dst,
                                 const float* __restrict__ b,
                                 float* __restrict__ out,
                                 int n_nodes) {
    int t = blockIdx.x * blockDim.x + threadIdx.x;
    if (t >= n_nodes * HIDDEN) return;
    int node = t >> 6;
    int feat = t & 63;
    float v = agg[t] * norm_dst[node] + b[feat];
    out[t] = fmaxf(v, 0.0f);
}

// ------------------------------------------------------------------ driver
extern "C" void kernel_launch(void* const* d_in, const int* in_sizes, int n_in,
                              void* d_out, int out_size, void* d_ws, size_t ws_size,
                              hipStream_t stream) {
    (void)in_sizes; (void)n_in; (void)out_size; (void)ws_size;

    const float* x  = (const float*)d_in[0];
    const float* W1 = (const float*)d_in[1];
    const float* b1 = (const float*)d_in[2];
    const float* W2 = (const float*)d_in[3];
    const float* b2 = (const float*)d_in[4];
    const int* src  = (const int*)d_in[5];
    const int* dst  = (const int*)d_in[6];
    float* out = (float*)d_out;

    float* ws       = (float*)d_ws;
    float* norm_src = ws;                                   // N_NODES
    float* norm_dst = ws + N_NODES;                         // N_NODES
    float* hbuf     = ws + 2 * (size_t)N_NODES;             // N_NODES * HIDDEN
    float* aggbuf   = hbuf + (size_t)N_NODES * HIDDEN;      // N_NODES * HIDDEN

    const int feat_elems = N_NODES * HIDDEN;                // 6,400,000

    // --- degrees -> norms ---
    zero_f32_kernel<<<(2 * N_NODES + 255) / 256, 256, 0, stream>>>(norm_src, 2 * N_NODES);
    degree_kernel<<<(N_EDGES + 255) / 256, 256, 0, stream>>>(src, dst, norm_src, norm_dst, N_EDGES);
    norm_finalize_kernel<<<(2 * N_NODES + 255) / 256, 256, 0, stream>>>(norm_src, 2 * N_NODES);

    const int tiles = (N_NODES / 16) * (HIDDEN / 16);       // 25,000 waves
    const int gemm_blocks = (tiles + 7) / 8;                // 8 waves / 256-thread block

    // --- layer 1: h1 = (x*norm_src)@W1 ; agg ; relu ---
    gemm_norm_wmma_kernel<IN_FEATS><<<gemm_blocks, 256, 0, stream>>>(x, W1, norm_src, hbuf, N_NODES);
    zero_f32_kernel<<<(feat_elems + 255) / 256, 256, 0, stream>>>(aggbuf, feat_elems);
    edge_scatter_kernel<<<(N_EDGES * 32 + 255) / 256, 256, 0, stream>>>(hbuf, src, dst, aggbuf, N_EDGES);
    bias_relu_kernel<<<(feat_elems + 255) / 256, 256, 0, stream>>>(aggbuf, norm_dst, b1, hbuf, N_NODES);

    // --- layer 2: h2 = (l1*norm_src)@W2 ; agg into d_out ; relu in place ---
    gemm_norm_wmma_kernel<HIDDEN><<<gemm_blocks, 256, 0, stream>>>(hbuf, W2, norm_src, aggbuf, N_NODES);
    zero_f32_kernel<<<(feat_elems + 255) / 256, 256, 0, stream>>>(out, feat_elems);
    edge_scatter_kernel<<<(N_EDGES * 32 + 255) / 256, 256, 0, stream>>>(aggbuf, src, dst, out, N_EDGES);
    bias_relu_kernel<<<(feat_elems + 255) / 256, 256, 0, stream>>>(out, norm_dst, b2, out, N_NODES);
}